// GCNEncoder_6107443495224
// MI455X (gfx1250) — compile-verified
//
#include <hip/hip_runtime.h>

#define N_NODES 50000
#define F_IN    5
#define F_H     128
#define F_OUT   64

typedef __attribute__((ext_vector_type(2))) float v2f;
typedef __attribute__((ext_vector_type(8))) float v8f;

// ---------------------------------------------------------------- utilities
__global__ void fill_kernel(float* __restrict__ p, int n, float v) {
    int i = blockIdx.x * blockDim.x + threadIdx.x;
    if (i < n) p[i] = v;
}

// deg[dst] += 1 for every edge (deg pre-seeded with 1.0 for the self-loop)
__global__ void deg_kernel(const int* __restrict__ dst, int E,
                           float* __restrict__ deg) {
    int e = blockIdx.x * blockDim.x + threadIdx.x;
    if (e < E) atomicAdd(&deg[dst[e]], 1.0f);
}

// dis[i] = rsqrt(deg[i])   (deg >= 1 always, self-loop included)
__global__ void rsqrt_kernel(float* __restrict__ d, int n) {
    int i = blockIdx.x * blockDim.x + threadIdx.x;
    if (i < n) d[i] = rsqrtf(d[i]);
}

// ------------------------------------------------- layer-1 GEMM via WMMA f32
// h1[N,128] = x[N,5] @ W1[5,128]; K padded 5 -> 8 (two 16x16x4 steps).
// Grid: 3125 blocks (M tiles), 256 threads = 8 waves = 8 N-tiles. Exact fit,
// EXEC all-ones at every WMMA. All loads unconditional (in-bounds for every
// lane); zero-padding done with value selects (v_cndmask), never EXEC.
__global__ void __launch_bounds__(256)
gemm1_wmma(const float* __restrict__ X, const float* __restrict__ W1,
           float* __restrict__ H1) {
    const int mt   = blockIdx.x;          // 0..3124
    const int nt   = threadIdx.x >> 5;    // 0..7  (N tile of 128)
    const int lane = threadIdx.x & 31;
    const int half = lane >> 4;           // 0: K=k0,k0+1   1: K=k0+2,k0+3
    const int l    = lane & 15;
    const int row  = mt * 16 + l;         // A row (M)
    const int col  = nt * 16 + l;         // B col (N)

    // unconditional loads (all in-bounds), issued up front
    const float a0 = X[row * F_IN + half * 2 + 0];
    const float a1 = X[row * F_IN + half * 2 + 1];
    const float a4 = X[row * F_IN + 4];
    const float b0 = W1[(half * 2 + 0) * F_H + col];
    const float b1v = W1[(half * 2 + 1) * F_H + col];
    const float b4 = W1[4 * F_H + col];

    v8f c = {};

    // --- K step 0: K = 0..3 (all valid)
    {
        v2f a, b;
        a.x = a0;  a.y = a1;
        b.x = b0;  b.y = b1v;
        c = __builtin_amdgcn_wmma_f32_16x16x4_f32(false, a, false, b,
                                                  (short)0, c, false, false);
    }
    // --- K step 1: K = 4..7 (only K=4 valid; pad via value select)
    {
        v2f a, b;
        a.x = (half == 0) ? a4 : 0.0f;
        a.y = 0.0f;
        b.x = (half == 0) ? b4 : 0.0f;
        b.y = 0.0f;
        c = __builtin_amdgcn_wmma_f32_16x16x4_f32(false, a, false, b,
                                                  (short)0, c, false, false);
    }

    // store D: VGPR r -> rows M=r (lanes 0-15) / M=r+8 (lanes 16-31)
    float* out = H1 + (mt * 16 + half * 8) * F_H + nt * 16 + l;
#pragma unroll
    for (int r = 0; r < 8; ++r) out[r * F_H] = c[r];
}

// -------------------------------------------- layer-1 edge aggregation
// agg1[dst] += dis[src]*dis[dst] * h1[src]; one wave per edge, float4 per lane
__global__ void agg1_kernel(const int* __restrict__ src,
                            const int* __restrict__ dst, int E,
                            const float* __restrict__ dis,
                            const float* __restrict__ h1,
                            float* __restrict__ agg1) {
    int t    = blockIdx.x * blockDim.x + threadIdx.x;
    int e    = t >> 5;
    int lane = t & 31;
    if (e >= E) return;
    int s = src[e], d = dst[e];
    float norm = dis[s] * dis[d];
    float4 v = ((const float4*)(h1 + (size_t)s * F_H))[lane];
    float* op = agg1 + (size_t)d * F_H + lane * 4;
    atomicAdd(op + 0, norm * v.x);
    atomicAdd(op + 1, norm * v.y);
    atomicAdd(op + 2, norm * v.z);
    atomicAdd(op + 3, norm * v.w);
}

// self-loop + bias + ReLU:  agg1 = relu(agg1 + dis^2*h1 + b1)
__global__ void post1_kernel(const float* __restrict__ dis,
                             const float* __restrict__ h1,
                             const float* __restrict__ b1,
                             float* __restrict__ agg1) {
    int i = blockIdx.x * blockDim.x + threadIdx.x;
    if (i >= N_NODES * F_H) return;
    int n = i / F_H, f = i % F_H;
    float di = dis[n];
    float v = agg1[i] + di * di * h1[i] + b1[f];
    agg1[i] = fmaxf(v, 0.0f);
}

// ------------------------------------------------- layer-2 GEMM via WMMA f32
// h2[N,64] = agg1[N,128] @ W2[128,64]; 32 K-steps of 16x16x4.
// Grid: 3125 blocks (M tiles), 128 threads = 4 waves = 4 N-tiles. Exact fit.
__global__ void __launch_bounds__(128)
gemm2_wmma(const float* __restrict__ A, const float* __restrict__ W2,
           float* __restrict__ H2) {
    const int mt   = blockIdx.x;          // 0..3124
    const int nt   = threadIdx.x >> 5;    // 0..3
    const int lane = threadIdx.x & 31;
    const int half = lane >> 4;
    const int l    = lane & 15;

    const float* aBase = A + ((size_t)(mt * 16 + l)) * F_H + half * 2;
    const float* bBase = W2 + (half * 2) * F_OUT + nt * 16 + l;

    v8f c = {};
#pragma unroll
    for (int k0 = 0; k0 < F_H; k0 += 4) {
        v2f a = *(const v2f*)(aBase + k0);       // A[row][k0+2h], A[row][k0+2h+1]
        v2f b;
        b.x = bBase[k0 * F_OUT];                 // W2[k0+2h  ][col]
        b.y = bBase[k0 * F_OUT + F_OUT];         // W2[k0+2h+1][col]
        c = __builtin_amdgcn_wmma_f32_16x16x4_f32(false, a, false, b,
                                                  (short)0, c, false, false);
    }

    float* out = H2 + (size_t)(mt * 16 + half * 8) * F_OUT + nt * 16 + l;
#pragma unroll
    for (int r = 0; r < 8; ++r) out[r * F_OUT] = c[r];
}

// -------------------------------------------- layer-2 edge aggregation
// out[dst] += dis[src]*dis[dst] * h2[src]; one wave per edge, float2 per lane
__global__ void agg2_kernel(const int* __restrict__ src,
                            const int* __restrict__ dst, int E,
                            const float* __restrict__ dis,
                            const float* __restrict__ h2,
                            float* __restrict__ out) {
    int t    = blockIdx.x * blockDim.x + threadIdx.x;
    int e    = t >> 5;
    int lane = t & 31;
    if (e >= E) return;
    int s = src[e], d = dst[e];
    float norm = dis[s] * dis[d];
    float2 v = ((const float2*)(h2 + (size_t)s * F_OUT))[lane];
    float* op = out + (size_t)d * F_OUT + lane * 2;
    atomicAdd(op + 0, norm * v.x);
    atomicAdd(op + 1, norm * v.y);
}

// self-loop + bias:  out += dis^2*h2 + b2
__global__ void post2_kernel(const float* __restrict__ dis,
                             const float* __restrict__ h2,
                             const float* __restrict__ b2,
                             float* __restrict__ out) {
    int i = blockIdx.x * blockDim.x + threadIdx.x;
    if (i >= N_NODES * F_OUT) return;
    int n = i / F_OUT, f = i % F_OUT;
    float di = dis[n];
    out[i] += di * di * h2[i] + b2[f];
}

// ---------------------------------------------------------------- launcher
extern "C" void kernel_launch(void* const* d_in, const int* in_sizes, int n_in,
                              void* d_out, int out_size, void* d_ws, size_t ws_size,
                              hipStream_t stream) {
    const float* x     = (const float*)d_in[0];
    const int*   edges = (const int*)d_in[1];   // [2, E] flat
    const float* W1    = (const float*)d_in[2];
    const float* b1    = (const float*)d_in[3];
    const float* W2    = (const float*)d_in[4];
    const float* b2    = (const float*)d_in[5];
    float* out = (float*)d_out;

    const int E = in_sizes[1] / 2;
    const int* srcv = edges;
    const int* dstv = edges + E;

    // workspace: dis[N] | buf[N*128] (h1, later reused as h2) | agg1[N*128]
    float* ws   = (float*)d_ws;
    float* dis  = ws;
    float* h1   = ws + N_NODES;
    float* agg1 = h1 + (size_t)N_NODES * F_H;
    float* h2   = h1;   // reuse: h1 dead after post1, h2 produced after

    const int T = 256;
    // init: deg=1 (self-loop), agg1=0, out=0
    fill_kernel<<<(N_NODES + T - 1) / T, T, 0, stream>>>(dis, N_NODES, 1.0f);
    fill_kernel<<<(N_NODES * F_H + T - 1) / T, T, 0, stream>>>(agg1, N_NODES * F_H, 0.0f);
    fill_kernel<<<(N_NODES * F_OUT + T - 1) / T, T, 0, stream>>>(out, N_NODES * F_OUT, 0.0f);

    // degree -> dis = rsqrt(deg)
    deg_kernel<<<(E + T - 1) / T, T, 0, stream>>>(dstv, E, dis);
    rsqrt_kernel<<<(N_NODES + T - 1) / T, T, 0, stream>>>(dis, N_NODES);

    // layer 1
    gemm1_wmma<<<N_NODES / 16, 256, 0, stream>>>(x, W1, h1);
    agg1_kernel<<<((E * 32) + T - 1) / T, T, 0, stream>>>(srcv, dstv, E, dis, h1, agg1);
    post1_kernel<<<(N_NODES * F_H + T - 1) / T, T, 0, stream>>>(dis, h1, b1, agg1);

    // layer 2
    gemm2_wmma<<<N_NODES / 16, 128, 0, stream>>>(agg1, W2, h2);
    agg2_kernel<<<((E * 32) + T - 1) / T, T, 0, stream>>>(srcv, dstv, E, dis, h2, out);
    post2_kernel<<<(N_NODES * F_OUT + T - 1) / T, T, 0, stream>>>(dis, h2, b2, out);
}